// GCN_Bin_2190433321550
// MI455X (gfx1250) — compile-verified
//
#include <hip/hip_runtime.h>
#include <hip/hip_bf16.h>

// CDNA5 / gfx1250 GCN (3-layer) — f32 pipeline.
// Heavy phase = L2-resident gather + global_atomic_add_f32 scatter over 1.6M edges.
// Dense feature transforms use V_WMMA_F32_16X16X4_F32 (wave32, 16x16 tile per wave).

typedef __attribute__((ext_vector_type(2))) float v2f;
typedef __attribute__((ext_vector_type(8))) float v8f;

#define TPB 256

__global__ void fill_f32(float* __restrict__ p, size_t n, float v) {
  size_t i = (size_t)blockIdx.x * blockDim.x + threadIdx.x;
  if (i < n) p[i] = v;
}

// degree over dst (self-loop handled by initializing deg=1.0)
__global__ void deg_count(const int* __restrict__ dst, float* __restrict__ deg, int E) {
  int e = blockIdx.x * blockDim.x + threadIdx.x;
  if (e < E) unsafeAtomicAdd(&deg[dst[e]], 1.0f);
}

__global__ void rsqrt_inplace(float* __restrict__ d, int n) {
  int i = blockIdx.x * blockDim.x + threadIdx.x;
  if (i < n) d[i] = rsqrtf(d[i]);  // deg >= 1 always (self loops)
}

// layer-1 transform: in_dim == 1 -> outer product x[i] * W1[f]
__global__ void transform1(const float* __restrict__ x, const float* __restrict__ W1,
                           float* __restrict__ t, int n) {
  long long tid = (long long)blockIdx.x * blockDim.x + threadIdx.x;
  if (tid < (long long)n * 32) {
    int i = (int)(tid >> 5);
    int f = (int)(tid & 31);
    t[tid] = x[i] * W1[f];
  }
}

// Dense transform T[N,OUTN] = H[N,IN] @ W[IN,OUTN] via V_WMMA_F32_16X16X4_F32.
// One wave per 16-row tile. A: lane m holds row (row0+m), VGPR0/1 = K+0/K+1,
// lane-half selects K+0..1 vs K+2..3. B mirrored. C/D: M = v + 8*half, N = lane&15.
template<int IN, int OUTN>
__global__ void gemm_wmma(const float* __restrict__ H, const float* __restrict__ W,
                          float* __restrict__ T, int nrows) {
  int wave = (int)((blockIdx.x * blockDim.x + threadIdx.x) >> 5);
  int lane = threadIdx.x & 31;
  int row0 = wave << 4;
  if (row0 >= nrows) return;          // uniform per wave: EXEC all-1s at WMMA

  int m    = lane & 15;
  int half = lane >> 4;               // 0: K lanes 0..1, 1: K lanes 2..3
  int n    = m;                       // output column for this lane

  int arow = row0 + m;
  if (arow > nrows - 1) arow = nrows - 1;   // partial-tile safety (N%16==0 here)

  v8f acc = {};
#pragma unroll
  for (int k = 0; k < IN; k += 4) {
    int kk = k + half * 2;
    const float2 av = *reinterpret_cast<const float2*>(H + (size_t)arow * IN + kk);
    v2f a; a[0] = av.x; a[1] = av.y;
    v2f b;
    b[0] = (n < OUTN) ? W[(size_t)kk * OUTN + n]       : 0.0f;
    b[1] = (n < OUTN) ? W[(size_t)(kk + 1) * OUTN + n] : 0.0f;
    // (neg_a, A, neg_b, B, c_mod, C, reuse_a, reuse_b)
    acc = __builtin_amdgcn_wmma_f32_16x16x4_f32(false, a, false, b, (short)0, acc,
                                                false, false);
  }

  if (n < OUTN) {
    if (row0 + 16 <= nrows) {
      // full tile (always the case when nrows % 16 == 0): one base address,
      // 8 stores with compile-time strides -> immediate-offset global_store_b32
      float* p = T + (size_t)(row0 + half * 8) * OUTN + n;
#pragma unroll
      for (int v = 0; v < 8; ++v) p[(size_t)v * OUTN] = acc[v];
    } else {
#pragma unroll
      for (int v = 0; v < 8; ++v) {
        int row = row0 + half * 8 + v;
        if (row < nrows) T[(size_t)row * OUTN + n] = acc[v];
      }
    }
  }
}

// message passing: agg[dst,f] += t[src,f] * dinv[src]*dinv[dst]
// vectorized: one thread per (edge, 4-feature chunk) when F % 4 == 0;
// gather is global_load_b128, scatter stays per-float global_atomic_add_f32.
template<int F>
__global__ void aggregate_vec4(const int* __restrict__ src, const int* __restrict__ dst,
                               const float* __restrict__ dinv, const float* __restrict__ t,
                               float* __restrict__ agg, int E) {
  constexpr int C = F / 4;
  long long tid = (long long)blockIdx.x * blockDim.x + threadIdx.x;
  if (tid >= (long long)E * C) return;
  int e = (int)(tid / C);
  int c = (int)(tid - (long long)e * C);
  int s = src[e], d = dst[e];
  float w = dinv[s] * dinv[d];
  const float4 v = *reinterpret_cast<const float4*>(t + (size_t)s * F + c * 4);
  float* o = agg + (size_t)d * F + c * 4;
  unsafeAtomicAdd(o + 0, v.x * w);
  unsafeAtomicAdd(o + 1, v.y * w);
  unsafeAtomicAdd(o + 2, v.z * w);
  unsafeAtomicAdd(o + 3, v.w * w);
}

// scalar fallback for F not divisible by 4 (layer 3, F = 11)
template<int F>
__global__ void aggregate(const int* __restrict__ src, const int* __restrict__ dst,
                          const float* __restrict__ dinv, const float* __restrict__ t,
                          float* __restrict__ agg, int E) {
  long long tid = (long long)blockIdx.x * blockDim.x + threadIdx.x;
  if (tid >= (long long)E * F) return;
  int e = (int)(tid / F);
  int f = (int)(tid - (long long)e * F);
  int s = src[e], d = dst[e];
  float w = dinv[s] * dinv[d];
  unsafeAtomicAdd(&agg[(size_t)d * F + f], t[(size_t)s * F + f] * w);
}

// out = agg + self_loop (t[i]*dinv[i]^2) + bias, optional ReLU
template<int F, bool RELU>
__global__ void finalize(const float* __restrict__ agg, const float* __restrict__ t,
                         const float* __restrict__ dinv, const float* __restrict__ bias,
                         float* __restrict__ out, int nrows) {
  long long tid = (long long)blockIdx.x * blockDim.x + threadIdx.x;
  if (tid >= (long long)nrows * F) return;
  int i = (int)(tid / F);
  int f = (int)(tid - (long long)i * F);
  float di = dinv[i];
  float v = agg[tid] + t[tid] * di * di + bias[f];
  out[tid] = RELU ? fmaxf(v, 0.0f) : v;
}

static inline unsigned nblk(long long total) {
  return (unsigned)((total + TPB - 1) / TPB);
}

extern "C" void kernel_launch(void* const* d_in, const int* in_sizes, int n_in,
                              void* d_out, int out_size, void* d_ws, size_t ws_size,
                              hipStream_t stream) {
  const float* x  = (const float*)d_in[0];
  const int*   ei = (const int*)d_in[1];     // edge_index [2,E]
  const float* W1 = (const float*)d_in[2];
  const float* b1 = (const float*)d_in[3];
  const float* W2 = (const float*)d_in[4];
  const float* b2 = (const float*)d_in[5];
  const float* W3 = (const float*)d_in[6];
  const float* b3 = (const float*)d_in[7];
  float* out = (float*)d_out;

  const int N = in_sizes[0];
  const int E = in_sizes[1] / 2;
  const int* src = ei;
  const int* dst = ei + E;

  // workspace layout (floats): dinv[N] | bufT[N*32] | bufA[N*32] | bufH[N*32]
  float* dinv = (float*)d_ws;
  float* bufT = dinv + N;
  float* bufA = bufT + (size_t)N * 32;
  float* bufH = bufA + (size_t)N * 32;

  // ---- normalization: deg (init 1.0 for self loop) -> dinv = rsqrt(deg)
  fill_f32<<<nblk(N), TPB, 0, stream>>>(dinv, (size_t)N, 1.0f);
  deg_count<<<nblk(E), TPB, 0, stream>>>(dst, dinv, E);
  rsqrt_inplace<<<nblk(N), TPB, 0, stream>>>(dinv, N);

  const int tiles = (N + 15) / 16;

  // ---- layer 1: F=32
  transform1<<<nblk((long long)N * 32), TPB, 0, stream>>>(x, W1, bufT, N);
  fill_f32<<<nblk((long long)N * 32), TPB, 0, stream>>>(bufA, (size_t)N * 32, 0.0f);
  aggregate_vec4<32><<<nblk((long long)E * 8), TPB, 0, stream>>>(src, dst, dinv, bufT, bufA, E);
  finalize<32, true><<<nblk((long long)N * 32), TPB, 0, stream>>>(bufA, bufT, dinv, b1, bufH, N);

  // ---- layer 2: [N,32] @ [32,16], F=16
  gemm_wmma<32, 16><<<nblk((long long)tiles * 32), TPB, 0, stream>>>(bufH, W2, bufT, N);
  fill_f32<<<nblk((long long)N * 16), TPB, 0, stream>>>(bufA, (size_t)N * 16, 0.0f);
  aggregate_vec4<16><<<nblk((long long)E * 4), TPB, 0, stream>>>(src, dst, dinv, bufT, bufA, E);
  finalize<16, true><<<nblk((long long)N * 16), TPB, 0, stream>>>(bufA, bufT, dinv, b2, bufH, N);

  // ---- layer 3: [N,16] @ [16,11], F=11, no ReLU, writes d_out
  gemm_wmma<16, 11><<<nblk((long long)tiles * 32), TPB, 0, stream>>>(bufH, W3, bufT, N);
  fill_f32<<<nblk((long long)N * 11), TPB, 0, stream>>>(bufA, (size_t)N * 11, 0.0f);
  aggregate<11><<<nblk((long long)E * 11), TPB, 0, stream>>>(src, dst, dinv, bufT, bufA, E);
  finalize<11, false><<<nblk((long long)N * 11), TPB, 0, stream>>>(bufA, bufT, dinv, b3, out, N);
}